// MultiHeadAttentionLayer_2594160247140
// MI455X (gfx1250) — compile-verified
//
#include <hip/hip_runtime.h>

// ---------------------------------------------------------------------------
// HGT-style multi-head attention layer on a relational graph, for MI455X
// (gfx1250, wave32). Dense projection GEMMs run on the fp32 WMMA path
// (V_WMMA_F32_16X16X4_F32) with the weight matrix staged transposed in LDS.
// ---------------------------------------------------------------------------

#define IN_DIM 128
#define H_HEADS 8
#define D_HEAD 16
#define R_RELS 9

typedef __attribute__((ext_vector_type(2))) float v2f;
typedef __attribute__((ext_vector_type(8))) float v8f;

// ---------------------------------------------------------------------------
// W[r][i][o] = sum_b comb[r][b] * basis[b][i][o]
// ---------------------------------------------------------------------------
__global__ __launch_bounds__(256)
void combine_bases(const float* __restrict__ basis, const float* __restrict__ comb,
                   float* __restrict__ Wout) {
    long idx = (long)blockIdx.x * 256 + threadIdx.x;
    const long total = (long)R_RELS * IN_DIM * IN_DIM;
    if (idx >= total) return;
    int r  = (int)(idx / (IN_DIM * IN_DIM));
    int io = (int)(idx % (IN_DIM * IN_DIM));
    float s = 0.0f;
#pragma unroll
    for (int b = 0; b < R_RELS; ++b)
        s += comb[r * R_RELS + b] * basis[(long)b * IN_DIM * IN_DIM + io];
    Wout[idx] = s;
}

// ---------------------------------------------------------------------------
// tmp[p][n][o] = sum_i h[n][i] * Wall[p][r][i][o]   (p = blockIdx.z in 0..2)
// Block: 256 threads = 8 waves; block tile = 128 rows x 128 cols.
// Wave w computes rows [128*bx + 16w, +16), all 128 cols: 8 WMMA col-tiles.
// W (128x128 fp32, 64 KB) is staged TRANSPOSED in LDS so the B fragment
// (rows k, k+1 of W for one column) is one ds_load_b64 per lane.
// ---------------------------------------------------------------------------
__global__ __launch_bounds__(256)
void gemm_xw(const float* __restrict__ h, const float* __restrict__ Wall,
             float* __restrict__ tmp, int r, int n_nodes) {
    __shared__ float Wt[IN_DIM * IN_DIM];   // Wt[o*128 + i] = W[i*128 + o]

    const float* W = Wall + ((long)blockIdx.z * R_RELS + r) * (IN_DIM * IN_DIM);
    float* out = tmp + (long)blockIdx.z * n_nodes * IN_DIM;

    const int t = threadIdx.x;
    // cooperative transposed load of W into LDS (64 elements per thread)
    for (int idx = t; idx < IN_DIM * IN_DIM; idx += 256) {
        int i = idx >> 7;
        int o = idx & 127;
        Wt[o * IN_DIM + i] = W[idx];
    }
    __syncthreads();

    const int wave    = t >> 5;        // 0..7
    const int lane    = t & 31;
    const int halfsel = lane >> 4;     // 0 or 1
    const int lm      = lane & 15;

    const long row0 = (long)blockIdx.x * 128 + wave * 16;

    // A fragment addressing (ISA 7.12.2): lane holds m=lm, k = 2*halfsel + {0,1}
    long arow = row0 + lm;
    if (arow >= n_nodes) arow = n_nodes - 1;   // clamp; padded rows never stored
    const float* aptr = h + arow * IN_DIM + 2 * halfsel;

    v8f acc[8];
#pragma unroll
    for (int i = 0; i < 8; ++i) acc[i] = (v8f)0.0f;

#pragma unroll 4
    for (int k0 = 0; k0 < IN_DIM; k0 += 4) {
        const v2f a = *(const v2f*)(aptr + k0);          // global_load_b64
        const int krow = k0 + 2 * halfsel;
#pragma unroll
        for (int nt = 0; nt < 8; ++nt) {
            // B fragment: VGPR v holds row (krow + v) of W for column ncol
            const int ncol = nt * 16 + lm;
            const v2f b = *(const v2f*)(&Wt[ncol * IN_DIM + krow]);  // ds_load_b64
            acc[nt] = __builtin_amdgcn_wmma_f32_16x16x4_f32(
                false, a, false, b, (short)0, acc[nt], false, false);
        }
    }

    // C/D layout: VGPR v, lanes 0-15 -> M=v, lanes 16-31 -> M=v+8; N = lm
#pragma unroll
    for (int nt = 0; nt < 8; ++nt) {
#pragma unroll
        for (int v = 0; v < 8; ++v) {
            const long rrow = row0 + v + 8 * halfsel;
            if (rrow < n_nodes)
                out[rrow * IN_DIM + nt * 16 + lm] = acc[nt][v];
        }
    }
}

// ---------------------------------------------------------------------------
// For edges with etype == r:  agg[p][dst] += tmp[p][src]   (p = Q,K,V)
// 2 edges per 256-thread block; thread handles one of 128 columns.
// ---------------------------------------------------------------------------
__global__ __launch_bounds__(256)
void scatter_edges(const float* __restrict__ tmp, float* __restrict__ agg,
                   const int* __restrict__ src, const int* __restrict__ dst,
                   const int* __restrict__ etype, int r, int E, int n_nodes) {
    long e = (long)blockIdx.x * 2 + (threadIdx.x >> 7);
    if (e >= E) return;
    if (etype[e] != r) return;
    const int c = threadIdx.x & 127;
    const long s = src[e];
    const long d = dst[e];
    const long stride = (long)n_nodes * IN_DIM;
#pragma unroll
    for (int p = 0; p < 3; ++p) {
        float v = tmp[p * stride + s * IN_DIM + c];
        atomicAdd(&agg[p * stride + d * IN_DIM + c], v);
    }
}

// ---------------------------------------------------------------------------
// agg[p][n][c] = relu(agg[p][n][c] + bias_p[c])
// ---------------------------------------------------------------------------
__global__ __launch_bounds__(256)
void bias_relu(float* __restrict__ agg, const float* __restrict__ Qb,
               const float* __restrict__ Kb, const float* __restrict__ Vb,
               int n_nodes) {
    long idx = (long)blockIdx.x * 256 + threadIdx.x;
    const long stride = (long)n_nodes * IN_DIM;
    if (idx >= 3 * stride) return;
    const int p = (int)(idx / stride);
    const int c = (int)(idx & (IN_DIM - 1));
    const float* b = (p == 0) ? Qb : ((p == 1) ? Kb : Vb);
    float v = agg[idx] + b[c];
    agg[idx] = v > 0.0f ? v : 0.0f;
}

// ---------------------------------------------------------------------------
// Per-edge attention: score = exp(clip(dot(K[src,h], Q[dst,h]) / 4, -10, 10))
// wV[dst,h,d] += V[src,h,d]*score ;  z[dst,h] += score
// 2 edges / 256-thread block; 16-lane subgroups reduce the per-head dot.
// ---------------------------------------------------------------------------
__global__ __launch_bounds__(256)
void attn_edges(const float* __restrict__ qkv, const int* __restrict__ src,
                const int* __restrict__ dst, float* __restrict__ wV,
                float* __restrict__ z, int E, int n_nodes) {
    long e = (long)blockIdx.x * 2 + (threadIdx.x >> 7);
    if (e >= E) return;
    const int t = threadIdx.x & 127;
    const int hd = t;                 // h*16 + d
    const int d  = t & 15;
    const int hh = t >> 4;
    const long stride = (long)n_nodes * IN_DIM;
    const float* Q = qkv;
    const float* K = qkv + stride;
    const float* V = qkv + 2 * stride;
    const long s  = src[e];
    const long dn = dst[e];

    float prod = K[s * IN_DIM + hd] * Q[dn * IN_DIM + hd];
#pragma unroll
    for (int off = 8; off >= 1; off >>= 1)
        prod += __shfl_xor(prod, off, 16);

    float sc = prod * 0.25f;                      // 1/sqrt(16)
    sc = fminf(10.0f, fmaxf(-10.0f, sc));
    sc = expf(sc);

    atomicAdd(&wV[dn * IN_DIM + hd], V[s * IN_DIM + hd] * sc);
    if (d == 0) atomicAdd(&z[dn * H_HEADS + hh], sc);
}

// ---------------------------------------------------------------------------
// out[n,h,d] /= (z[n,h] + 1e-6)
// ---------------------------------------------------------------------------
__global__ __launch_bounds__(256)
void normalize_out(float* __restrict__ out, const float* __restrict__ z,
                   int n_nodes) {
    long idx = (long)blockIdx.x * 256 + threadIdx.x;
    if (idx >= (long)n_nodes * IN_DIM) return;
    const long n = idx >> 7;
    const int hh = (int)((idx >> 4) & (H_HEADS - 1));
    out[idx] = out[idx] / (z[n * H_HEADS + hh] + 1e-6f);
}

// ---------------------------------------------------------------------------
// Host-side launch
// ---------------------------------------------------------------------------
extern "C" void kernel_launch(void* const* d_in, const int* in_sizes, int n_in,
                              void* d_out, int out_size, void* d_ws, size_t ws_size,
                              hipStream_t stream) {
    const float* h   = (const float*)d_in[0];
    const float* Qw  = (const float*)d_in[1];
    const float* Qc  = (const float*)d_in[2];
    const float* Qb  = (const float*)d_in[3];
    const float* Kw  = (const float*)d_in[4];
    const float* Kc  = (const float*)d_in[5];
    const float* Kb  = (const float*)d_in[6];
    const float* Vw  = (const float*)d_in[7];
    const float* Vc  = (const float*)d_in[8];
    const float* Vb  = (const float*)d_in[9];
    const int*   src = (const int*)d_in[10];
    const int*   dst = (const int*)d_in[11];
    const int*   ety = (const int*)d_in[12];
    float* out = (float*)d_out;

    const int nN = in_sizes[0] / IN_DIM;   // 50000
    const int E  = in_sizes[10];           // 800000

    // workspace layout (256B-aligned chunks)
    char* ws = (char*)d_ws;
    size_t off = 0;
    auto carve = [&](size_t bytes) {
        size_t p = off;
        off += (bytes + 255) & ~(size_t)255;
        return p;
    };
    const size_t wmat_elems = 3ull * R_RELS * IN_DIM * IN_DIM;   // 27 weight mats
    const size_t proj_elems = 3ull * (size_t)nN * IN_DIM;        // Q|K|V planes
    float* Wall = (float*)(ws + carve(wmat_elems * sizeof(float)));
    float* tmp  = (float*)(ws + carve(proj_elems * sizeof(float)));
    float* agg  = (float*)(ws + carve(proj_elems * sizeof(float)));
    float* z    = (float*)(ws + carve((size_t)nN * H_HEADS * sizeof(float)));
    (void)ws_size;

    // zero accumulators (harness poisons ws/out; graph-capture-safe async memset)
    hipMemsetAsync(agg, 0, proj_elems * sizeof(float), stream);
    hipMemsetAsync(z, 0, (size_t)nN * H_HEADS * sizeof(float), stream);
    hipMemsetAsync(out, 0, (size_t)out_size * sizeof(float), stream);

    // 1) basis combination for Q, K, V weight stacks
    const int cgrid = (int)(((long)R_RELS * IN_DIM * IN_DIM + 255) / 256);
    combine_bases<<<cgrid, 256, 0, stream>>>(Qw, Qc, Wall);
    combine_bases<<<cgrid, 256, 0, stream>>>(Kw, Kc, Wall + R_RELS * IN_DIM * IN_DIM);
    combine_bases<<<cgrid, 256, 0, stream>>>(Vw, Vc, Wall + 2 * R_RELS * IN_DIM * IN_DIM);

    // 2) per relation: WMMA projection GEMMs then edge scatter-accumulate
    dim3 ggrid((nN + 127) / 128, 1, 3);
    const int sgrid = (E + 1) / 2;
    for (int r = 0; r < R_RELS; ++r) {
        gemm_xw<<<ggrid, 256, 0, stream>>>(h, Wall, tmp, r, nN);
        scatter_edges<<<sgrid, 256, 0, stream>>>(tmp, agg, src, dst, ety, r, E, nN);
    }

    // 3) bias + relu (agg becomes final Q|K|V)
    const long be = 3l * nN * IN_DIM;
    bias_relu<<<(unsigned)((be + 255) / 256), 256, 0, stream>>>(agg, Qb, Kb, Vb, nN);

    // 4) edge attention with atomic segment sums
    attn_edges<<<sgrid, 256, 0, stream>>>(agg, src, dst, out, z, E, nN);

    // 5) normalize
    const long oe = (long)nN * IN_DIM;
    normalize_out<<<(unsigned)((oe + 255) / 256), 256, 0, stream>>>(out, z, nN);
}